// VersatileConvSE3_51110110822701
// MI455X (gfx1250) — compile-verified
//
#include <hip/hip_runtime.h>
#include <stdint.h>

#define E_TOTAL 10000
#define CI      16
#define CO      16
#define FREQ    44
#define IN_DIM  16
#define OUT_DIM 16
#define EDGE_F  16
#define MID     32
#define LN_EPS  1e-5f

#define EB      16            // edges per block
#define NWAVES  16
#define THREADS (NWAVES * 32)
#define RW_ROW  17            // padded ci-stride
#define RW_EDGE (CO * RW_ROW) // 272 floats per edge row

// basis streaming: FC freqs per chunk, async-staged to LDS, triple buffered
#define FC      2
#define NC      (FREQ / FC)                 // 22 chunks
#define ROW_DW  40                          // padded dwords per (in) row (160B, 16B aligned)
#define EDGE_CHUNK_DW (IN_DIM * ROW_DW)     // 640 dwords per edge per chunk

typedef float v2f __attribute__((ext_vector_type(2)));
typedef float v8f __attribute__((ext_vector_type(8)));

__device__ __forceinline__ v8f wmma_f32(v2f a, v2f b, v8f c) {
    return __builtin_amdgcn_wmma_f32_16x16x4_f32(false, a, false, b, (short)0, c,
                                                 false, false);
}

// Async-stage one (edge, f-chunk) basis tile [16 in][FC*16 out] into LDS.
// Each lane moves 16B; 4 x GLOBAL_LOAD_ASYNC_TO_LDS_B128 covers 16 rows x 128B.
__device__ __forceinline__ void prefetch_chunk(const float* bs_edge, int f0,
                                               float* dst, int lane) {
    const int r4 = lane >> 3;          // row group 0..3
    const int c4 = (lane & 7) * 4;     // float column within 32-float row
    #pragma unroll
    for (int i = 0; i < 4; ++i) {
        const int in = i * 4 + r4;
        const uint64_t g = (uint64_t)(bs_edge + (size_t)in * (FREQ * OUT_DIM)
                                              + (size_t)f0 * OUT_DIM + c4);
        // low 32 bits of a flat LDS pointer == LDS byte offset
        const uint32_t l = (uint32_t)(uintptr_t)(dst + in * ROW_DW + c4);
        asm volatile("global_load_async_to_lds_b128 %0, %1, off"
                     :: "v"(l), "v"(g) : "memory");
    }
}

__global__ __launch_bounds__(THREADS)
void conv_se3_fused(const float* __restrict__ features,
                    const float* __restrict__ ief,
                    const float* __restrict__ basis,
                    const float* __restrict__ w1, const float* __restrict__ b1,
                    const float* __restrict__ g1, const float* __restrict__ be1,
                    const float* __restrict__ w2, const float* __restrict__ b2,
                    const float* __restrict__ g2, const float* __restrict__ be2,
                    const float* __restrict__ w3,
                    float* __restrict__ out)
{
    __shared__ float feat_lds[EB * CI * IN_DIM];                  // 16 KB
    __shared__ float h1_lds[EB * MID];                            //  2 KB (wave-private)
    __shared__ float h2_lds[EB * 33];                             //  padded stride 33
    __shared__ float rw_lds[2][EB * RW_EDGE];                     //  double-buffered RW
    __shared__ float tmp_lds[NWAVES][CI * OUT_DIM];               //  D->B relayout buffer
    __shared__ __align__(16) float bas_lds[3][NWAVES * EDGE_CHUNK_DW]; // 120 KB basis pipe

    const int tid  = threadIdx.x;
    const int wave = tid >> 5;          // edge (stages 0/2/3), co-tile (stage 1)
    const int lane = tid & 31;
    const int hl   = lane & 15;
    const int kh   = lane >> 4;
    const int e0   = blockIdx.x * EB;
    if (e0 >= E_TOTAL) return;

    // ---------------- stage features tile into LDS (coalesced) ----------------------
    for (int i = tid; i < EB * CI * IN_DIM; i += THREADS)
        feat_lds[i] = features[(size_t)e0 * CI * IN_DIM + i];

    // ---------------- kick off async basis pipeline (chunks 0,1) --------------------
    const float* bs_edge = basis + (size_t)(e0 + wave) * IN_DIM * FREQ * OUT_DIM;
    prefetch_chunk(bs_edge, 0 * FC, &bas_lds[0][wave * EDGE_CHUNK_DW], lane);
    prefetch_chunk(bs_edge, 1 * FC, &bas_lds[1][wave * EDGE_CHUNK_DW], lane);

    // ---------------- RadialProfile MLP (wave = edge, lane = unit) ------------------
    {
        const int e = e0 + wave;
        const int m = lane;
        float s = b1[m];
        #pragma unroll
        for (int k = 0; k < EDGE_F; ++k)
            s += ief[e * EDGE_F + k] * w1[m * EDGE_F + k];
        float sum = s, sq = s * s;
        #pragma unroll
        for (int d = 16; d >= 1; d >>= 1) { sum += __shfl_xor(sum, d); sq += __shfl_xor(sq, d); }
        float mean = sum * (1.0f / MID);
        float var  = sq  * (1.0f / MID) - mean * mean;
        float h1 = fmaxf((s - mean) * rsqrtf(var + LN_EPS) * g1[m] + be1[m], 0.0f);
        h1_lds[wave * MID + m] = h1;

        float s2 = b2[m];
        #pragma unroll
        for (int k = 0; k < MID; ++k)
            s2 += h1_lds[wave * MID + k] * w2[m * MID + k];
        sum = s2; sq = s2 * s2;
        #pragma unroll
        for (int d = 16; d >= 1; d >>= 1) { sum += __shfl_xor(sum, d); sq += __shfl_xor(sq, d); }
        mean = sum * (1.0f / MID);
        var  = sq  * (1.0f / MID) - mean * mean;
        float h2 = fmaxf((s2 - mean) * rsqrtf(var + LN_EPS) * g2[m] + be2[m], 0.0f);
        h2_lds[wave * 33 + m] = h2;
    }
    __syncthreads();

    v8f acc = {};   // out[co, o] accumulator for edge 'wave' (D-layout)

    for (int c = 0; c < NC; ++c) {
        if (c + 2 < NC)  // keep 2 chunks in flight
            prefetch_chunk(bs_edge, (c + 2) * FC,
                           &bas_lds[(c + 2) % 3][wave * EDGE_CHUNK_DW], lane);
        const float* bch = &bas_lds[c % 3][wave * EDGE_CHUNK_DW];

        #pragma unroll
        for (int fc = 0; fc < FC; ++fc) {
            const int f = c * FC + fc;
            float* rw = rw_lds[f & 1];

            // ---- stage 1: RW_f tile for co = wave, all 16 edges (M=e, N=ci, K=32) --
            v8f d1 = {};
            #pragma unroll
            for (int ks = 0; ks < 8; ++ks) {
                const int m0 = ks * 4 + kh * 2;
                v2f a; a.x = h2_lds[hl * 33 + m0];
                       a.y = h2_lds[hl * 33 + m0 + 1];
                const float* wp = w3 + ((size_t)(wave * 16 + hl) * FREQ + f) * MID + m0;
                v2f b; b.x = wp[0]; b.y = wp[1];
                d1 = wmma_f32(a, b, d1);
            }
            #pragma unroll
            for (int v = 0; v < 8; ++v)
                rw[(v + 8 * kh) * RW_EDGE + wave * RW_ROW + hl] = d1[v];

            // chunk c's async tile must be resident before first B-fragment read
            if (fc == 0) {
                if (c + 2 < NC)      asm volatile("s_wait_asynccnt 0x8" ::: "memory");
                else if (c + 1 < NC) asm volatile("s_wait_asynccnt 0x4" ::: "memory");
                else                 asm volatile("s_wait_asynccnt 0x0" ::: "memory");
            }

            // ---- stage 2: TMP_f = features(16x16) @ basis_f(16x16), edge = wave ----
            v8f t2 = {};
            #pragma unroll
            for (int ks = 0; ks < 4; ++ks) {
                const int in0 = ks * 4 + kh * 2;
                v2f a; a.x = feat_lds[wave * 256 + hl * 16 + in0];
                       a.y = feat_lds[wave * 256 + hl * 16 + in0 + 1];
                v2f b; b.x = bch[in0 * ROW_DW + fc * OUT_DIM + hl];
                       b.y = bch[(in0 + 1) * ROW_DW + fc * OUT_DIM + hl];
                t2 = wmma_f32(a, b, t2);
            }
            #pragma unroll
            for (int v = 0; v < 8; ++v)
                tmp_lds[wave][(v + 8 * kh) * 16 + hl] = t2[v];

            __syncthreads();   // RW_f complete; fences reads of rw_lds[(f-1)&1]

            // ---- stage 3: acc += RW_f(co x ci) @ TMP_f(ci x o) ---------------------
            #pragma unroll
            for (int ks = 0; ks < 4; ++ks) {
                const int c0 = ks * 4 + kh * 2;
                v2f a; a.x = rw[wave * RW_EDGE + hl * RW_ROW + c0];
                       a.y = rw[wave * RW_EDGE + hl * RW_ROW + c0 + 1];
                v2f b; b.x = tmp_lds[wave][c0 * 16 + hl];
                       b.y = tmp_lds[wave][(c0 + 1) * 16 + hl];
                acc = wmma_f32(a, b, acc);
            }
        }
    }

    // ---------------- epilogue: out[e, co, o], streaming store -----------------------
    float* op = out + (size_t)(e0 + wave) * CO * OUT_DIM;
    #pragma unroll
    for (int v = 0; v < 8; ++v)
        __builtin_nontemporal_store(acc[v], op + (v + 8 * kh) * 16 + hl);
}

extern "C" void kernel_launch(void* const* d_in, const int* in_sizes, int n_in,
                              void* d_out, int out_size, void* d_ws, size_t ws_size,
                              hipStream_t stream) {
    const float* features = (const float*)d_in[0];
    const float* ief      = (const float*)d_in[1];
    const float* basis    = (const float*)d_in[2];
    const float* w1       = (const float*)d_in[3];
    const float* b1       = (const float*)d_in[4];
    const float* g1       = (const float*)d_in[5];
    const float* be1      = (const float*)d_in[6];
    const float* w2       = (const float*)d_in[7];
    const float* b2       = (const float*)d_in[8];
    const float* g2       = (const float*)d_in[9];
    const float* be2      = (const float*)d_in[10];
    const float* w3       = (const float*)d_in[11];
    float* out            = (float*)d_out;

    dim3 grid((E_TOTAL + EB - 1) / EB);
    dim3 block(THREADS);
    hipLaunchKernelGGL(conv_se3_fused, grid, block, 0, stream,
                       features, ief, basis, w1, b1, g1, be1, w2, b2, g2, be2, w3, out);
}